// AdExNeuron_3642132267481
// MI455X (gfx1250) — compile-verified
//
#include <hip/hip_runtime.h>
#include <stdint.h>

// ---------------------------------------------------------------------------
// AdEx neuron scan for MI455X (gfx1250).
//
// Strictly sequential nonlinear recurrence -> single wave32 owns the chain.
// Critical path per step:  v_fma -> v_exp_f32 -> v_fma (+cndmask reset).
// Data movement (32MB in / 32MB out, trivial vs 23.3 TB/s) is hidden via the
// Tensor Data Mover: double-buffered TENSOR_LOAD_TO_LDS of 4KB chunks,
// synchronized with s_wait_tensorcnt. Input broadcast per step via
// v_readlane_b32; spike outputs gathered one-per-lane -> coalesced stores.
// ---------------------------------------------------------------------------

typedef unsigned int u32x4 __attribute__((ext_vector_type(4)));
typedef int          i32x8 __attribute__((ext_vector_type(8)));
typedef int          i32x4 __attribute__((ext_vector_type(4)));

#define CHUNK 1024                  // floats per TDM chunk (4 KB)
#define LOG2E 1.44269504088896340736f

// Build D# (CDNA5 ISA) for a 1-D tile of `rem` f32 elements
// starting at gsrc, landing at LDS byte offset lds_byte, and issue the DMA.
// This toolchain's builtin is the 6-arg (clang-23 / therock-10.0) form.
__device__ __forceinline__ void tdm_load_chunk(const float* gsrc,
                                               uint32_t lds_byte,
                                               uint32_t rem) {
  uint64_t ga = (uint64_t)(uintptr_t)gsrc;
  // Group 0: count=1 | lds_addr | global_addr[56:0] | type=2 ("image")
  u32x4 g0 = { 1u,
               lds_byte,
               (uint32_t)(ga & 0xFFFFFFFFu),
               (uint32_t)((ga >> 32) & 0x1FFFFFFu) | (2u << 30) };
  // Group 1: data_size=2 (4B). tensor_dim0=rem (bits 79:48), tensor_dim1=1,
  // tile_dim0=rem (bits 127:112), tile_dim1/2 unused, dim0_stride=rem.
  i32x8 g1 = { (int)(2u << 16),
               (int)((rem & 0xFFFFu) << 16),
               (int)(((rem >> 16) & 0xFFFFu) | (1u << 16)),
               (int)((rem & 0xFFFFu) << 16),
               0,
               (int)rem,
               0,
               0 };
  i32x4 g2 = { 0, 0, 0, 0 };
  i32x4 g3 = { 0, 0, 0, 0 };
  i32x8 gx = { 0, 0, 0, 0, 0, 0, 0, 0 };
  __builtin_amdgcn_tensor_load_to_lds(g0, g1, g2, g3, gx, 0);
}

__global__ void adex_scan_kernel(const float* __restrict__ I,
                                 float* __restrict__ out,
                                 int n,
                                 const float* pC,  const float* pgL,
                                 const float* pEL, const float* pVT,
                                 const float* pdT, const float* pa,
                                 const float* pb,  const float* ptauw,
                                 const float* pVr) {
  extern __shared__ float lds_dyn[];          // 2 * CHUNK floats (8 KB)
  const int lane = (int)threadIdx.x;          // wave32: 0..31

  const float C    = *pC;
  const float gL   = *pgL;
  const float EL   = *pEL;
  const float VT   = *pVT;
  const float dT   = *pdT;
  const float a    = *pa;
  const float b    = *pb;
  const float tauw = *ptauw;
  const float Vr   = *pVr;

  // Fused per-step constants (DT == 1):
  // v' = A*v + B0 + E*exp2(c3*v + c4) - G*w + G*I
  // w' = Wc*w + Wa*v + Wb
  const float G  = 1.0f / C;
  const float A  = 1.0f - gL * G;
  const float B0 = gL * EL * G;
  const float E  = gL * dT * G;
  const float c3 = LOG2E / dT;
  const float c4 = -VT * LOG2E / dT;
  const float Wc = 1.0f - 1.0f / tauw;
  const float Wa = a / tauw;
  const float Wb = -a * EL / tauw;

  const int steps   = n - 1;                      // spikes[1..n-1]
  const int nchunks = (n + CHUNK - 1) / CHUNK;
  const uint32_t lds_base = __builtin_amdgcn_groupstaticsize(); // dyn-LDS offset

  if (lane == 0) out[0] = 0.0f;                   // spikes[0] == 0 by definition

  // Prologue: prime the double buffer two chunks deep.
  {
    uint32_t rem0 = (uint32_t)((n < CHUNK) ? n : CHUNK);
    tdm_load_chunk(I, lds_base, rem0);
    if (nchunks > 1) {
      int left = n - CHUNK;
      uint32_t rem1 = (uint32_t)((left < CHUNK) ? left : CHUNK);
      tdm_load_chunk(I + CHUNK, lds_base + CHUNK * 4u, rem1);
    }
  }

  float v = EL;
  float w = 0.0f;

  for (int c = 0; c < nchunks; ++c) {
    // Chunks issued so far: min(c+2, nchunks); need chunk c resident.
    if (c + 1 < nchunks) __builtin_amdgcn_s_wait_tensorcnt(1);
    else                 __builtin_amdgcn_s_wait_tensorcnt(0);
    asm volatile("" ::: "memory");   // keep ds_loads below the wait

    const int bsel = c & 1;
    const float* lb = &lds_dyn[bsel * CHUNK];
    const int base = c * CHUNK;

    for (int gi = 0; gi < CHUNK / 32; ++gi) {
      // Coalesced, conflict-free LDS pull: lane L holds I[base+gi*32+L].
      float Iv = lb[gi * 32 + lane];
      float sp = 0.0f;

#pragma unroll
      for (int k = 0; k < 32; ++k) {
        // Broadcast lane k's input to the whole wave (v_readlane_b32, imm k).
        float It = __uint_as_float(
            __builtin_amdgcn_readlane(__float_as_uint(Iv), k));

        // Independent side chain for t; critical chain is fma->exp2->fma.
        float t  = __builtin_fmaf(A, v, B0);
        t        = __builtin_fmaf(-G, w, t);
        t        = __builtin_fmaf(G, It, t);
        float ex = __builtin_amdgcn_exp2f(__builtin_fmaf(c3, v, c4));
        float vn = __builtin_fmaf(E, ex, t);
        float wn = __builtin_fmaf(Wc, w, __builtin_fmaf(Wa, v, Wb));

        bool spike = (vn >= 30.0f);               // VCUT
        v = spike ? Vr : vn;
        w = wn + (spike ? b : 0.0f);
        if (lane == k) sp = spike ? 1.0f : 0.0f;  // lane k archives step k
      }

      int sidx = base + gi * 32 + lane;           // step index of this lane
      if (sidx < steps) out[sidx + 1] = sp;       // coalesced 128B store
    }

    asm volatile("" ::: "memory");
    // Refill the buffer we just drained with chunk c+2.
    int nc = c + 2;
    if (nc < nchunks) {
      int left = n - nc * CHUNK;
      uint32_t rem = (uint32_t)((left < CHUNK) ? left : CHUNK);
      tdm_load_chunk(I + (size_t)nc * CHUNK,
                     lds_base + (uint32_t)bsel * (CHUNK * 4u), rem);
    }
  }
}

extern "C" void kernel_launch(void* const* d_in, const int* in_sizes, int n_in,
                              void* d_out, int out_size, void* d_ws, size_t ws_size,
                              hipStream_t stream) {
  (void)n_in; (void)out_size; (void)d_ws; (void)ws_size;
  const float* I    = (const float*)d_in[0];
  const float* pC   = (const float*)d_in[1];
  const float* pgL  = (const float*)d_in[2];
  const float* pEL  = (const float*)d_in[3];
  const float* pVT  = (const float*)d_in[4];
  const float* pdT  = (const float*)d_in[5];
  const float* pa   = (const float*)d_in[6];
  const float* pb   = (const float*)d_in[7];
  const float* ptau = (const float*)d_in[8];
  const float* pVr  = (const float*)d_in[9];
  float* out = (float*)d_out;
  int n = in_sizes[0];

  adex_scan_kernel<<<1, 32, 2 * CHUNK * sizeof(float), stream>>>(
      I, out, n, pC, pgL, pEL, pVT, pdT, pa, pb, ptau, pVr);
}